// Net_53919019434016
// MI455X (gfx1250) — compile-verified
//
#include <hip/hip_runtime.h>
#include <math.h>

#define N_NODES   100000
#define N_EDGES   3200000
#define NFEAT     128
#define DIM       32
#define NCLASS    10
#define NGRAPHS   512
#define BN_EPS    1e-5f

typedef float v2f __attribute__((ext_vector_type(2)));
typedef float v8f __attribute__((ext_vector_type(8)));

// ---------------------------------------------------------------- zero fill
__global__ void zero_kernel(float4* __restrict__ p, int n4) {
  int i = blockIdx.x * blockDim.x + threadIdx.x;
  if (i < n4) p[i] = make_float4(0.f, 0.f, 0.f, 0.f);
}

// --------------------------------------------- edge scatter-add (segment_sum)
// thread -> (edge, float4-chunk); native global_atomic_add_f32 via unsafeAtomicAdd
template <int SHIFT>  // chunks per edge = 1<<SHIFT  (features = 4<<SHIFT)
__global__ void scatter_add_kernel(const float* __restrict__ x,
                                   const int* __restrict__ src,
                                   const int* __restrict__ dst,
                                   float* __restrict__ agg) {
  long tid = (long)blockIdx.x * blockDim.x + threadIdx.x;
  const long total = (long)N_EDGES << SHIFT;
  if (tid >= total) return;
  const int e = (int)(tid >> SHIFT);
  const int c = (int)(tid & ((1 << SHIFT) - 1));
  const int F = 4 << SHIFT;
  const int s = src[e], d = dst[e];
  const float4 v = ((const float4*)(x + (long)s * F))[c];
  float* base = agg + (long)d * F + (c << 2);
  unsafeAtomicAdd(base + 0, v.x);
  unsafeAtomicAdd(base + 1, v.y);
  unsafeAtomicAdd(base + 2, v.z);
  unsafeAtomicAdd(base + 3, v.w);
}

// --------------------------------------- fused GIN MLP + BN, fp32 WMMA 16x16x4
// One wave (wave32) processes one 16-node tile. Two N-tiles (cols 0-15,16-31).
template <int FIN>
__global__ __launch_bounds__(128)
void gin_mlp_bn_kernel(const float* __restrict__ xin,   // [N, FIN]
                       const float* __restrict__ agg,   // [N, FIN]
                       const float* __restrict__ W1,    // [FIN, 32]
                       const float* __restrict__ b1,
                       const float* __restrict__ W2,    // [32, 32]
                       const float* __restrict__ b2,
                       const float* __restrict__ gam,
                       const float* __restrict__ bet,
                       const float* __restrict__ mu,
                       const float* __restrict__ var,
                       float* __restrict__ hout,        // [N, 32]
                       int n_tiles) {
  constexpr int LD1 = FIN + 2;   // padded leading dims -> conflict-free ds_load_b64
  constexpr int LDM = DIM + 2;   // 34
  __shared__ float sW1t[DIM * LD1];     // W1 transposed: [n][k]
  __shared__ float sW2t[DIM * LDM];     // W2 transposed: [n][k]
  __shared__ float sMid[4][16 * LDM];   // per-wave mid activation staging

  const int t = threadIdx.x;
  for (int i = t; i < DIM * FIN; i += 128) {
    const int n = i / FIN, k = i % FIN;
    sW1t[n * LD1 + k] = W1[k * DIM + n];
  }
  for (int i = t; i < DIM * DIM; i += 128) {
    const int n = i / DIM, k = i % DIM;
    sW2t[n * LDM + k] = W2[k * DIM + n];
  }
  __syncthreads();

  const int wave = t >> 5;
  const int lane = t & 31;
  const int tile = blockIdx.x * 4 + wave;
  const bool active = tile < n_tiles;         // uniform per wave
  const int m    = lane & 15;                 // M-row (A) / N-col (B,C)
  const int kh   = (lane >> 4) << 1;          // K offset within chunk: 0 or 2
  const int mrow = (lane >> 4) << 3;          // C-layout row base: 0 or 8
  float* mid = sMid[wave];

  if (active) {
    const long row0 = (long)tile * 16;
    v8f c0 = {}; v8f c1 = {};
    const float* xr = xin + (row0 + m) * FIN;
    const float* ar = agg + (row0 + m) * FIN;
#pragma unroll 4
    for (int k = 0; k < FIN; k += 4) {
      const int kr = k + kh;
      const float2 xv = *(const float2*)(xr + kr);
      const float2 av = *(const float2*)(ar + kr);
      v2f a; a.x = xv.x + av.x; a.y = xv.y + av.y;     // h = x + agg (GIN eps=0)
      v2f bA, bB;
      bA.x = sW1t[m * LD1 + kr];          bA.y = sW1t[m * LD1 + kr + 1];
      bB.x = sW1t[(m + 16) * LD1 + kr];   bB.y = sW1t[(m + 16) * LD1 + kr + 1];
      c0 = __builtin_amdgcn_wmma_f32_16x16x4_f32(false, a, false, bA, (short)0, c0, false, false);
      c1 = __builtin_amdgcn_wmma_f32_16x16x4_f32(false, a, false, bB, (short)0, c1, false, false);
    }
    // bias + ReLU, stage to LDS row-major for GEMM2 A-operand re-layout
    const float bb0 = b1[m], bb1 = b1[m + 16];
#pragma unroll
    for (int r = 0; r < 8; ++r) {
      mid[(mrow + r) * LDM + m]      = fmaxf(c0[r] + bb0, 0.f);
      mid[(mrow + r) * LDM + m + 16] = fmaxf(c1[r] + bb1, 0.f);
    }
  }
  __syncthreads();

  if (active) {
    const long row0 = (long)tile * 16;
    v8f d0 = {}; v8f d1 = {};
#pragma unroll
    for (int k = 0; k < DIM; k += 4) {
      const int kr = k + kh;
      v2f a; a.x = mid[m * LDM + kr]; a.y = mid[m * LDM + kr + 1];
      v2f bA, bB;
      bA.x = sW2t[m * LDM + kr];          bA.y = sW2t[m * LDM + kr + 1];
      bB.x = sW2t[(m + 16) * LDM + kr];   bB.y = sW2t[(m + 16) * LDM + kr + 1];
      d0 = __builtin_amdgcn_wmma_f32_16x16x4_f32(false, a, false, bA, (short)0, d0, false, false);
      d1 = __builtin_amdgcn_wmma_f32_16x16x4_f32(false, a, false, bB, (short)0, d1, false, false);
    }
    // bias + ReLU + BatchNorm (eval): y = s*h + t,  s=gamma*rsqrt(var+eps), t=beta-mean*s
    const float s0 = gam[m]      * rsqrtf(var[m]      + BN_EPS);
    const float s1 = gam[m + 16] * rsqrtf(var[m + 16] + BN_EPS);
    const float t0 = bet[m]      - mu[m]      * s0;
    const float t1 = bet[m + 16] - mu[m + 16] * s1;
    const float bb0 = b2[m], bb1 = b2[m + 16];
#pragma unroll
    for (int r = 0; r < 8; ++r) {
      const float h0 = fmaxf(d0[r] + bb0, 0.f);
      const float h1 = fmaxf(d1[r] + bb1, 0.f);
      const long row = row0 + mrow + r;
      hout[row * DIM + m]      = fmaf(h0, s0, t0);
      hout[row * DIM + m + 16] = fmaf(h1, s1, t1);
    }
  }
}

// ------------------------------------------------------- global_add_pool
__global__ void pool_kernel(const float* __restrict__ h,
                            const int* __restrict__ batch,
                            float* __restrict__ g) {
  int tid = blockIdx.x * blockDim.x + threadIdx.x;
  if (tid >= N_NODES * 8) return;
  const int i = tid >> 3, c = tid & 7;
  const int b = batch[i];
  const float4 v = ((const float4*)(h + (long)i * DIM))[c];
  float* base = g + b * DIM + (c << 2);
  unsafeAtomicAdd(base + 0, v.x);
  unsafeAtomicAdd(base + 1, v.y);
  unsafeAtomicAdd(base + 2, v.z);
  unsafeAtomicAdd(base + 3, v.w);
}

// ---------------------------------------- FC head + log_softmax (512 rows)
__global__ void head_kernel(const float* __restrict__ g,
                            const float* __restrict__ fc1W, const float* __restrict__ fc1b,
                            const float* __restrict__ fc2W, const float* __restrict__ fc2b,
                            float* __restrict__ out) {
  const int gi = blockIdx.x * blockDim.x + threadIdx.x;
  if (gi >= NGRAPHS) return;
  float gr[DIM];
#pragma unroll
  for (int k = 0; k < DIM; ++k) gr[k] = g[gi * DIM + k];
  float tbuf[DIM];
  for (int j = 0; j < DIM; ++j) {
    float acc = fc1b[j];
#pragma unroll
    for (int k = 0; k < DIM; ++k) acc = fmaf(gr[k], fc1W[k * DIM + j], acc);
    tbuf[j] = fmaxf(acc, 0.f);
  }
  float l[NCLASS]; float mx = -INFINITY;
  for (int j = 0; j < NCLASS; ++j) {
    float acc = fc2b[j];
#pragma unroll
    for (int k = 0; k < DIM; ++k) acc = fmaf(tbuf[k], fc2W[k * NCLASS + j], acc);
    l[j] = acc; mx = fmaxf(mx, acc);
  }
  float s = 0.f;
  for (int j = 0; j < NCLASS; ++j) s += expf(l[j] - mx);
  const float lse = mx + logf(s);
  for (int j = 0; j < NCLASS; ++j) out[gi * NCLASS + j] = l[j] - lse;
}

// ---------------------------------------------------------------- launcher
extern "C" void kernel_launch(void* const* d_in, const int* in_sizes, int n_in,
                              void* d_out, int out_size, void* d_ws, size_t ws_size,
                              hipStream_t stream) {
  const float *x, *W1[5], *B1[5], *W2[5], *B2[5], *GAM[5], *BET[5], *MU[5], *VAR[5];
  const float *fc1W, *fc1b, *fc2W, *fc2b;
  const int *ei, *batch;
  auto F = [&](int i) { return (const float*)d_in[i]; };

  if (in_sizes[0] == N_NODES * NFEAT) {
    // dict insertion order: x, edge_index, batch, convs{W1,b1,W2,b2}x5, bns{gamma,beta,mean,var}x5, fc...
    x = F(0); ei = (const int*)d_in[1]; batch = (const int*)d_in[2];
    for (int l = 0; l < 5; ++l) {
      W1[l] = F(3 + 4 * l);  B1[l]  = F(4 + 4 * l);  W2[l] = F(5 + 4 * l);  B2[l]  = F(6 + 4 * l);
      GAM[l] = F(23 + 4 * l); BET[l] = F(24 + 4 * l); MU[l] = F(25 + 4 * l); VAR[l] = F(26 + 4 * l);
    }
    fc1W = F(43); fc1b = F(44); fc2W = F(45); fc2b = F(46);
  } else {
    // sorted pytree-leaf order: batch, edge_index, bns{beta,gamma,mean,var}x5, convs{W1,W2,b1,b2}x5, fc.., x
    batch = (const int*)d_in[0]; ei = (const int*)d_in[1];
    for (int l = 0; l < 5; ++l) {
      BET[l] = F(2 + 4 * l);  GAM[l] = F(3 + 4 * l);  MU[l] = F(4 + 4 * l);  VAR[l] = F(5 + 4 * l);
      W1[l]  = F(22 + 4 * l); W2[l]  = F(23 + 4 * l); B1[l] = F(24 + 4 * l); B2[l]  = F(25 + 4 * l);
    }
    fc1W = F(42); fc1b = F(43); fc2W = F(44); fc2b = F(45); x = F(46);
  }
  const int* src = ei;
  const int* dst = ei + N_EDGES;

  char* ws = (char*)d_ws;
  float* agg = (float*)ws;                                     // 51.2 MB (layer0)
  float* hA  = (float*)(ws + 51200000);                        // 12.8 MB
  float* hB  = (float*)(ws + 51200000 + 12800000);             // 12.8 MB
  float* g   = (float*)(ws + 51200000 + 2 * 12800000);         // 64 KB

  const int NTILES = N_NODES / 16;       // 6250, exact
  const int MLPBLK = (NTILES + 3) / 4;

  // ---- layer 0 (FIN = 128)
  {
    const int n4 = N_NODES * NFEAT / 4;
    zero_kernel<<<(n4 + 255) / 256, 256, 0, stream>>>((float4*)agg, n4);
    const long total = (long)N_EDGES << 5;
    scatter_add_kernel<5><<<(int)((total + 255) / 256), 256, 0, stream>>>(x, src, dst, agg);
    gin_mlp_bn_kernel<NFEAT><<<MLPBLK, 128, 0, stream>>>(
        x, agg, W1[0], B1[0], W2[0], B2[0], GAM[0], BET[0], MU[0], VAR[0], hA, NTILES);
  }
  // ---- layers 1..4 (FIN = 32)
  const float* cur = hA; float* nxt = hB;
  for (int l = 1; l < 5; ++l) {
    const int n4 = N_NODES * DIM / 4;
    zero_kernel<<<(n4 + 255) / 256, 256, 0, stream>>>((float4*)agg, n4);
    const long total = (long)N_EDGES << 3;
    scatter_add_kernel<3><<<(int)((total + 255) / 256), 256, 0, stream>>>(cur, src, dst, agg);
    gin_mlp_bn_kernel<DIM><<<MLPBLK, 128, 0, stream>>>(
        cur, agg, W1[l], B1[l], W2[l], B2[l], GAM[l], BET[l], MU[l], VAR[l], nxt, NTILES);
    float* t2 = (float*)cur; cur = nxt; nxt = t2;
  }
  // ---- pool + head
  {
    const int n4 = NGRAPHS * DIM / 4;
    zero_kernel<<<(n4 + 255) / 256, 256, 0, stream>>>((float4*)g, n4);
    const int total = N_NODES * 8;
    pool_kernel<<<(total + 255) / 256, 256, 0, stream>>>(cur, batch, g);
  }
  head_kernel<<<(NGRAPHS + 255) / 256, 256, 0, stream>>>(g, fc1W, fc1b, fc2W, fc2b, (float*)d_out);
}